// SquaredErrorLoss_26929444946237
// MI455X (gfx1250) — compile-verified
//
#include <hip/hip_runtime.h>
#include <hip/hip_bf16.h>

// Problem constants (from reference)
#define BB 4
#define NCH 3
#define NCLS 20
#define HH 512
#define WW 512
#define MAXB 50
#define ROWS 8                 // rows per chunk
#define NCHUNK (HH / ROWS)     // 64
#define NBOX (BB * MAXB)       // 200

typedef __attribute__((ext_vector_type(2))) float v2f;
typedef __attribute__((ext_vector_type(8))) float v8f;

// Wave-level reduce helper: 'val' per lane (wave32). Returns full wave sum in lane 0
// (valid in all lanes 0..15 group after shuffles). Uses V_WMMA_F32_16X16X4_F32 with
// A = ones(16x4), B = [val in rows K0/K1 halves, zeros in K2/K3] so
// D[m][n] = val[n] + val[n+16] (column sums, replicated over all m since A is ones).
__device__ __forceinline__ float wave_reduce_wmma(float val) {
    v2f a; a.x = 1.0f; a.y = 1.0f;          // ones A matrix (16x4 f32, 2 VGPRs)
    v2f b; b.x = val;  b.y = 0.0f;          // lanes 0-15 / 16-31 fill two K rows
    v8f c = {};
    c = __builtin_amdgcn_wmma_f32_16x16x4_f32(
        /*neg_a=*/false, a, /*neg_b=*/false, b,
        /*c_mod=*/(short)0, c, /*reuse_a=*/false, /*reuse_b=*/false);
    float t = c[0];                          // colsum_{lane%16} in every lane
    t += __shfl_xor(t, 1, 32);
    t += __shfl_xor(t, 2, 32);
    t += __shfl_xor(t, 4, 32);
    t += __shfl_xor(t, 8, 32);
    return t;                                // lane 0: sum of 16 distinct column sums
}

// Phase 1: one block per (batch*box, row-chunk). Accumulate sum of squared error
// over the box's pixels in this 8-row slab for the box's class channels.
__global__ void __launch_bounds__(256)
box_partial_kernel(const float* __restrict__ yfcn, const float* __restrict__ im,
                   const int* __restrict__ boxes, const int* __restrict__ nbox,
                   float* __restrict__ partial) {
    const int bj    = blockIdx.x;       // 0..199
    const int chunk = blockIdx.y;       // 0..63
    const int b = bj / MAXB, j = bj % MAXB;
    const int tid = threadIdx.x;

    const int* g = boxes + bj * 5;
    const int x1 = g[0], y1 = g[1], x2 = g[2], y2 = g[3], cls = g[4];
    const bool valid = (j < nbox[b]);

    const int r0 = chunk * ROWS;
    const int ys = (y1 > r0) ? y1 : r0;
    int ye = r0 + ROWS; if (y2 < ye) ye = y2;

    float acc = 0.0f;
    if (valid && ys < ye && x1 < x2) {
        const float* im_b = im   + (size_t)b * NCH  * HH * WW;
        const float* yf_b = yfcn + ((size_t)b * NCLS + cls) * NCH * HH * WW;
        for (int y = ys; y < ye; ++y) {
            const float* imr = im_b + (size_t)y * WW;
            const float* yfr = yf_b + (size_t)y * WW;
            if (y + 1 < ye) {   // stream-prefetch next row of the big y_fcn array
                __builtin_prefetch(yfr + WW + x1 + tid, 0, 0);
            }
            for (int x = x1 + tid; x < x2; x += 256) {
#pragma unroll
                for (int n = 0; n < NCH; ++n) {
                    const size_t co = (size_t)n * HH * WW;
                    float d = imr[co + x] - yfr[co + x];
                    acc = fmaf(d, d, acc);
                }
            }
        }
    }

    // Block reduction: 256 -> 32 via LDS, 32 -> 1 via WMMA + shuffles.
    __shared__ float red[256];
    red[tid] = acc;
    __syncthreads();
    if (tid < 32) {                       // exactly wave 0, EXEC all-ones for WMMA
        float s = 0.0f;
#pragma unroll
        for (int k = 0; k < 8; ++k) s += red[tid + 32 * k];
        float t = wave_reduce_wmma(s);
        if (tid == 0) partial[bj * NCHUNK + chunk] = t;
    }
}

// Phase 2: gather per-box chunk partials, normalize, mask invalid, average.
__global__ void __launch_bounds__(256)
finalize_kernel(const float* __restrict__ partial, const int* __restrict__ boxes,
                const int* __restrict__ nbox, float* __restrict__ out) {
    const int tid = threadIdx.x;
    float err = 0.0f;
    if (tid < NBOX) {
        const int b = tid / MAXB, j = tid % MAXB;
        if (j < nbox[b]) {
            float s = 0.0f;
            for (int k = 0; k < NCHUNK; ++k) s += partial[tid * NCHUNK + k];
            const int* g = boxes + tid * 5;
            int area = (g[3] - g[1]) * (g[2] - g[0]);
            if (area < 1) area = 1;
            err = s / ((float)NCH * (float)area);
        }
    }
    __shared__ float red[256];
    red[tid] = err;
    __syncthreads();
    if (tid < 32) {
        float s = 0.0f;
#pragma unroll
        for (int k = 0; k < 8; ++k) s += red[tid + 32 * k];
        float t = wave_reduce_wmma(s);
        if (tid == 0) {
            float tot = (float)(nbox[0] + nbox[1] + nbox[2] + nbox[3]);
            out[0] = t / tot;
        }
    }
}

extern "C" void kernel_launch(void* const* d_in, const int* in_sizes, int n_in,
                              void* d_out, int out_size, void* d_ws, size_t ws_size,
                              hipStream_t stream) {
    const float* y_fcn     = (const float*)d_in[0];
    const float* im_data   = (const float*)d_in[1];
    // d_in[2] = im_info (unused by reference math)
    const int*   gt_boxes  = (const int*)d_in[3];
    const int*   num_boxes = (const int*)d_in[4];
    float* out     = (float*)d_out;
    float* partial = (float*)d_ws;   // NBOX*NCHUNK floats = 51.2 KB

    dim3 grid(NBOX, NCHUNK);
    box_partial_kernel<<<grid, 256, 0, stream>>>(y_fcn, im_data, gt_boxes, num_boxes, partial);
    finalize_kernel<<<1, 256, 0, stream>>>(partial, gt_boxes, num_boxes, out);
}